// UndoMaxPooling2D_72000831750265
// MI455X (gfx1250) — compile-verified
//
#include <hip/hip_runtime.h>
#include <hip/hip_bf16.h>

// Max-unpooling scatter for MI455X (gfx1250).
//
// Pass 1: zero the 102.8 MB output with b128 stores (RT temporal hint so the
//         lines stay resident in the 192 MB L2).
// Pass 2: coalesced b128 loads of x/pos, then 4 random b32 scatter stores per
//         thread; the random stores hit the L2-resident output lines.

__global__ __launch_bounds__(256) void unpool_zero_kernel(float4* __restrict__ out,
                                                          int n4) {
    int i = blockIdx.x * blockDim.x + threadIdx.x;
    if (i < n4) {
        out[i] = make_float4(0.0f, 0.0f, 0.0f, 0.0f);  // global_store_b128, RT hint
    }
}

__global__ __launch_bounds__(256) void unpool_scatter_kernel(const float4* __restrict__ x4,
                                                             const int4* __restrict__ p4,
                                                             float* __restrict__ out,
                                                             int n4) {
    int i = blockIdx.x * blockDim.x + threadIdx.x;
    if (i < n4) {
        float4 v = x4[i];   // global_load_b128, coalesced
        int4   p = p4[i];   // global_load_b128, coalesced
        // Random scatter; output fits in L2 so these are L2 hits.
        out[p.x] = v.x;
        out[p.y] = v.y;
        out[p.z] = v.z;
        out[p.w] = v.w;
    }
}

extern "C" void kernel_launch(void* const* d_in, const int* in_sizes, int n_in,
                              void* d_out, int out_size, void* d_ws, size_t ws_size,
                              hipStream_t stream) {
    (void)d_ws; (void)ws_size; (void)n_in;

    const float* x   = (const float*)d_in[0];
    const int*   pos = (const int*)d_in[1];
    float*       out = (float*)d_out;

    const int n_in_elems = in_sizes[0];      // 6,422,528  (divisible by 4)
    const int n_out4     = out_size / 4;     // 6,422,528 float4 slots
    const int n_in4      = n_in_elems / 4;   // 1,605,632

    const int threads = 256;                 // 8 wave32 waves per block

    // Pass 1: zero-fill output (d_out is poisoned before timing; must zero every call).
    int zero_blocks = (n_out4 + threads - 1) / threads;
    unpool_zero_kernel<<<zero_blocks, threads, 0, stream>>>((float4*)out, n_out4);

    // Pass 2: scatter (same stream -> ordered after zero-fill).
    int scat_blocks = (n_in4 + threads - 1) / threads;
    unpool_scatter_kernel<<<scat_blocks, threads, 0, stream>>>(
        (const float4*)x, (const int4*)pos, out, n_in4);
}